// SPD_30167850287641
// MI455X (gfx1250) — compile-verified
//
#include <hip/hip_runtime.h>

// out[b] = expm(0.5 * v_hat[b])   (sqrtm(expm(X)) == expm(X/2) for symmetric X)
// Scaling-and-squaring with Paterson-Stockmeyer:
//   A  = 0.5*v_hat / 4                       (s = 2 scaling)
//   E  = W0 + A^4*(W1 + A^4*W2)              (degree-12 Taylor, 5 matmuls)
//   P  = E^2^2                               (2 squarings)
// Two 8x8 matrices packed block-diagonally per 16x16 WMMA tile; all matmuls use
// V_WMMA_F32_16X16X4_F32 (4 chained K-chunks per 16x16x16 product). The +W1/+W0
// adds ride for free in the WMMA C operand. Grid is capped so each wave loops
// over ~12 pairs, amortizing basis staging; next pair's v is prefetched.

typedef __attribute__((ext_vector_type(8))) float v8f;
typedef __attribute__((ext_vector_type(2))) float v2f;

#define DIM 36
#define WPB 8          // waves per 256-thread block
#define MAX_BLOCKS 2048

// 16x16x16 fp32 matmul: D = A*B + C, A/B given as 8 layout-VGPRs (4 chunks x 2)
__device__ __forceinline__ v8f mm16(const float A[8], const float B[8], v8f C) {
  v8f acc = C;
#pragma unroll
  for (int c = 0; c < 4; ++c) {
    v2f a, b;
    a[0] = A[2 * c]; a[1] = A[2 * c + 1];
    b[0] = B[2 * c]; b[1] = B[2 * c + 1];
    acc = __builtin_amdgcn_wmma_f32_16x16x4_f32(false, a, false, b, (short)0,
                                                acc, false, false);
  }
  return acc;
}

// Convert a SYMMETRIC 16x16 matrix from C/D layout (VGPR r, lane l ->
// (m=r+8*(l>=16), n=l%16)) to the A/B operand layout of wmma_f32_16x16x4
// (chunk c, reg j, lane l -> element (l%16, 4c+j+2*(l>=16))).
// For symmetric X, A-layout(X) == B-layout(X).
__device__ __forceinline__ void d_to_ab(v8f D, float L[8], bool lo) {
  float x0 = __shfl_xor((float)D[0], 16, 32);
  float x1 = __shfl_xor((float)D[1], 16, 32);
  float x2 = __shfl_xor((float)D[2], 16, 32);
  float x3 = __shfl_xor((float)D[3], 16, 32);
  float x4 = __shfl_xor((float)D[4], 16, 32);
  float x5 = __shfl_xor((float)D[5], 16, 32);
  float x6 = __shfl_xor((float)D[6], 16, 32);
  float x7 = __shfl_xor((float)D[7], 16, 32);
  L[0] = lo ? (float)D[0] : x2;  // k = 0 / 2
  L[1] = lo ? (float)D[1] : x3;  // k = 1 / 3
  L[2] = lo ? (float)D[4] : x6;  // k = 4 / 6
  L[3] = lo ? (float)D[5] : x7;  // k = 5 / 7
  L[4] = lo ? x0 : (float)D[2];  // k = 8 / 10
  L[5] = lo ? x1 : (float)D[3];  // k = 9 / 11
  L[6] = lo ? x4 : (float)D[6];  // k = 12 / 14
  L[7] = lo ? x5 : (float)D[7];  // k = 13 / 15
}

__global__ __launch_bounds__(256) void spd_expm_wmma(
    const float* __restrict__ v, const float* __restrict__ basis,
    float* __restrict__ out, int batch) {
  __shared__ float sb[DIM * 64];
  __shared__ float sv[WPB][2 * DIM];

  for (int i = threadIdx.x; i < DIM * 64; i += blockDim.x) sb[i] = basis[i];
  __syncthreads();

  const int lane = threadIdx.x & 31;
  const int wave = threadIdx.x >> 5;
  const bool lo = lane < 16;
  const bool act = (lane < 8) || (lane >= 24);  // on-block-diagonal lanes
  const int nloc = lo ? (lane & 7) : ((lane - 16) & 7);
  const float* vv = &sv[wave][lo ? 0 : DIM];

  // Block-diagonal identity in D-layout (zero in off-diagonal blocks).
  v8f Id;
#pragma unroll
  for (int r = 0; r < 8; ++r) {
    const int m = r + (lo ? 0 : 8);
    Id[r] = ((lane & 15) == m) ? 1.0f : 0.0f;
  }

  // Per-lane sparse basis gather, hoisted out of the pair loop: element (r,n)
  // of v_hat touches exactly one basis coefficient.
  int dIdx[8];
  float bc[8];
#pragma unroll
  for (int r = 0; r < 8; ++r) {
    const int i = (r < nloc) ? r : nloc;
    const int j = (r < nloc) ? nloc : r;
    const int d = (r == nloc) ? r : (8 + (i * (15 - i)) / 2 + (j - i - 1));
    dIdx[r] = d;
    // 0.5 (sqrt) * 1/4 (s=2 scaling), zeroed on off-diagonal-block lanes
    bc[r] = act ? sb[d * 64 + r * 8 + nloc] * 0.125f : 0.0f;
  }

  // 1/k! Taylor coefficients, k = 2..12
  const float c2 = 5.0000000e-01f, c3 = 1.6666667e-01f, c4 = 4.1666667e-02f,
              c5 = 8.3333333e-03f, c6 = 1.3888889e-03f, c7 = 1.9841270e-04f,
              c8 = 2.4801587e-05f, c9 = 2.7557319e-06f, c10 = 2.7557319e-07f,
              c11 = 2.5052108e-08f, c12 = 2.0876757e-09f;

  const int npairs = (batch + 1) >> 1;
  const int stride = gridDim.x * WPB;
  for (int pair = blockIdx.x * WPB + wave; pair < npairs; pair += stride) {
    const int b0 = pair * 2;
    const int b1 = b0 + 1;
    const bool has1 = (b1 < batch);
    const int bl1 = has1 ? b1 : b0;

    // Prefetch next iteration's tangent coords (288 B -> 3 cachelines).
    const int npair = pair + stride;
    if (npair < npairs) {
      __builtin_prefetch(&v[(size_t)npair * 2 * DIM + lane * 9], 0, 3);
    }

    // Stage this wave's 2x36 tangent coords into LDS (coalesced read).
    for (int i = lane; i < 2 * DIM; i += 32) {
      const int b = (i < DIM) ? b0 : bl1;
      const int d = (i < DIM) ? i : i - DIM;
      sv[wave][i] = v[(size_t)b * DIM + d];
    }

    // A = (0.5/4) * v_hat in D-layout via the 1-sparse basis gather.
    v8f A;
#pragma unroll
    for (int r = 0; r < 8; ++r) A[r] = vv[dIdx[r]] * bc[r];

    const v8f z = {};

    // Powers of A (all symmetric polynomials of A).
    float Aab[8];
    d_to_ab(A, Aab, lo);
    v8f A2 = mm16(Aab, Aab, z);
    float A2ab[8];
    d_to_ab(A2, A2ab, lo);
    v8f A3 = mm16(A2ab, Aab, z);
    v8f A4 = mm16(A2ab, A2ab, z);
    float A4ab[8];
    d_to_ab(A4, A4ab, lo);

    // Paterson-Stockmeyer groups (cheap D-layout linear combinations).
    v8f W0, W1, W2;
#pragma unroll
    for (int r = 0; r < 8; ++r) {
      W2[r] = c8 * Id[r] + c9 * A[r] + c10 * A2[r] + c11 * A3[r] + c12 * A4[r];
      W1[r] = c4 * Id[r] + c5 * A[r] + c6 * A2[r] + c7 * A3[r];
      W0[r] = Id[r] + A[r] + c2 * A2[r] + c3 * A3[r];
    }

    float Wab[8];
    d_to_ab(W2, Wab, lo);
    v8f T = mm16(A4ab, Wab, W1);  // W1 + A^4 * W2
    d_to_ab(T, Wab, lo);
    v8f P = mm16(A4ab, Wab, W0);  // W0 + A^4 * T   == exp(A) to degree 12

    // Undo the 2^-2 scaling: square twice (P symmetric each step).
#pragma unroll
    for (int sq = 0; sq < 2; ++sq) {
      float Pab[8];
      d_to_ab(P, Pab, lo);
      P = mm16(Pab, Pab, z);
    }

    // Store: lane l<8 holds column l of matrix b0 (rows r=0..7 in P[r]);
    // lane>=24 holds column lane-24 of matrix b1.
    if (lane < 8) {
      float* o = out + (size_t)b0 * 64 + lane;
#pragma unroll
      for (int r = 0; r < 8; ++r) o[r * 8] = P[r];
    } else if (lane >= 24 && has1) {
      float* o = out + (size_t)b1 * 64 + (lane - 24);
#pragma unroll
      for (int r = 0; r < 8; ++r) o[r * 8] = P[r];
    }
  }
}

extern "C" void kernel_launch(void* const* d_in, const int* in_sizes, int n_in,
                              void* d_out, int out_size, void* d_ws,
                              size_t ws_size, hipStream_t stream) {
  const float* v = (const float*)d_in[0];
  const float* basis = (const float*)d_in[1];
  float* out = (float*)d_out;
  const int batch = in_sizes[0] / DIM;
  const int npairs = (batch + 1) / 2;
  int blocks = (npairs + WPB - 1) / WPB;
  if (blocks > MAX_BLOCKS) blocks = MAX_BLOCKS;
  if (blocks < 1) blocks = 1;
  spd_expm_wmma<<<dim3(blocks), dim3(256), 0, stream>>>(v, basis, out, batch);
}